// DiffusionBlock_31018253811737
// MI455X (gfx1250) — compile-verified
//
#include <hip/hip_runtime.h>

typedef __attribute__((ext_vector_type(2))) float v2f;
typedef __attribute__((ext_vector_type(8))) float v8f;

#define IMG   1024
#define TILE  64
#define FUSE  4            // diffusion steps fused per kernel launch
#define ROWS0 (TILE + 2*FUSE)  // 72: input tile rows (halo 4)
#define P     84           // LDS pitch (floats), both buffers.
                           // 84 mod 64 = 20: column gathers -> banks {20M mod 64} (16 distinct,
                           // all =0 mod 4); +2-col half-wave -> =2 mod 4 (disjoint).
                           // 2*84 mod 64 = 40: row-gather half-waves in disjoint bank ranges.
                           // Even pitch => 8B-aligned b64 inner stores.

// Generic inner step: stage-k point (r,c) reads stage-(k-1) rows r..r+2, cols c..c+2.
// Processes 2 columns per iteration (shared loads, b64 store).
template <int N>
__device__ __forceinline__ void inner_step(const float* __restrict__ sA,
                                           float* __restrict__ sB,
                                           float hx, float hy, float cc) {
  constexpr int NH = N / 2;
  for (int idx = threadIdx.x; idx < N * NH; idx += 256) {
    int r = idx / NH;
    int c = (idx - r * NH) * 2;
    const float* a = &sA[r * P + c];
    float up0 = a[1],         up1 = a[2];
    float dn0 = a[2 * P + 1], dn1 = a[2 * P + 2];
    float l0  = a[P + 0], m0 = a[P + 1], m1 = a[P + 2], r1 = a[P + 3];
    float2 o;
    o.x = cc * m0 + hy * (up0 + dn0) + hx * (l0 + m1);
    o.y = cc * m1 + hy * (up1 + dn1) + hx * (m0 + r1);
    *(float2*)&sB[r * P + c] = o;          // c even, P even -> 8B aligned
  }
  __syncthreads();
}

// Four fused diffusion steps per launch:
//   async-LDS stage 72x72 (reflect halo 4) -> 3 scalar steps (70,68,66) -> WMMA final step.
// Final step: out = Ay*B + B*Ax with tridiagonal band matrices via V_WMMA_F32_16X16X4_F32.
__global__ __launch_bounds__(256) void diffusion_fused4_wmma(
    const float* __restrict__ src, const float* __restrict__ wgt,
    float* __restrict__ dst)
{
  __shared__ float buf0[ROWS0 * P];
  __shared__ float buf1[ROWS0 * P];

  const int plane = blockIdx.z;        // n*2 + c
  const int ch    = plane & 1;
  const int R0    = blockIdx.y * TILE;
  const int C0    = blockIdx.x * TILE;

  const float* s = src + (size_t)plane * (IMG * IMG);
  float*       d = dst + (size_t)plane * (IMG * IMG);

  // OIHW (2,1,3,3): hx = w[c,0,1,0], hy = w[c,0,0,1], cc = w[c,0,1,1]
  const float hx = wgt[ch * 9 + 3];
  const float hy = wgt[ch * 9 + 1];
  const float cc = wgt[ch * 9 + 4];

  // ---- stage 72x72 input tile (halo 4, reflect) straight into LDS (ASYNCcnt path) ----
  for (int idx = threadIdx.x; idx < ROWS0 * ROWS0; idx += 256) {
    int r = idx / ROWS0;
    int c = idx - r * ROWS0;
    int gr = R0 + r - FUSE;
    int gc = C0 + c - FUSE;
    gr = (gr < 0) ? -gr : ((gr >= IMG) ? (2 * IMG - 2 - gr) : gr);   // reflect (edge excl.)
    gc = (gc < 0) ? -gc : ((gc >= IMG) ? (2 * IMG - 2 - gc) : gc);
    unsigned lds_off = (unsigned)(size_t)&buf0[r * P + c];
    unsigned long long ga = (unsigned long long)(size_t)(s + (size_t)gr * IMG + gc);
    asm volatile("global_load_async_to_lds_b32 %0, %1, off"
                 :: "v"(lds_off), "v"(ga) : "memory");
  }
  asm volatile("s_wait_asynccnt 0" ::: "memory");
  __syncthreads();

  // ---- inner scalar steps (reflect borders emerge exactly from symmetric stencil) ----
  inner_step<70>(buf0, buf1, hx, hy, cc);   // row 0 == global R0-3
  inner_step<68>(buf1, buf0, hx, hy, cc);   // row 0 == global R0-2
  inner_step<66>(buf0, buf1, hx, hy, cc);   // row 0 == global R0-1

  const float* B = buf1;                    // 66x66, row 0 == global row R0-1

  // ---- final step (WMMA): out = Ay x B + B x Ax per 16x16 tile ----
  const int wave = threadIdx.x >> 5;
  const int lane = threadIdx.x & 31;
  const int nh   = lane >> 4;   // half-wave
  const int ln   = lane & 15;

  // Constant band matrices in WMMA operand layouts.
  // A-side (Ay[M][K], M=ln): off-diag hy, diag cc.  B-side (Ax[K][N], N=ln): off-diag hx.
  float wA[8], wB[8];
#pragma unroll
  for (int g = 0; g < 8; ++g) {
    int K  = 4 * (g >> 1) + (g & 1) + 2 * nh;
    int dd = ln - K;
    int off = (dd == 1) | (dd == -1);
    wA[g] = off ? hy : ((dd == 0) ? cc : 0.0f);
    wB[g] = off ? hx : 0.0f;
  }

  // 8 waves x 2 tiles = 16 tiles of 16x16 covering the 64x64 block
#pragma unroll
  for (int tt = 0; tt < 2; ++tt) {
    int t   = wave * 2 + tt;
    int ttr = (t >> 2) * 16;
    int ttc = (t & 3) * 16;

    float hb[8], ha[8];
#pragma unroll
    for (int g = 0; g < 8; ++g) {
      int K = 4 * (g >> 1) + (g & 1) + 2 * nh;
      hb[g] = B[(1 + ttr + K)  * P + (1 + ttc + ln)]; // B layout: H[K][N=ln]
      ha[g] = B[(1 + ttr + ln) * P + (1 + ttc + K)];  // A layout: H[M=ln][K]
    }

    v8f acc = {0.f, 0.f, 0.f, 0.f, 0.f, 0.f, 0.f, 0.f};
#pragma unroll
    for (int q = 0; q < 4; ++q) {   // vertical: acc += Ay x H
      v2f a = { wA[2 * q], wA[2 * q + 1] };
      v2f b = { hb[2 * q], hb[2 * q + 1] };
      acc = __builtin_amdgcn_wmma_f32_16x16x4_f32(false, a, false, b,
                                                  (short)0, acc, false, false);
    }
#pragma unroll
    for (int q = 0; q < 4; ++q) {   // horizontal: acc += H x Ax
      v2f a = { ha[2 * q], ha[2 * q + 1] };
      v2f b = { wB[2 * q], wB[2 * q + 1] };
      acc = __builtin_amdgcn_wmma_f32_16x16x4_f32(false, a, false, b,
                                                  (short)0, acc, false, false);
    }

    // halo fixups: neighbors just outside the 16x16 window (from B's halo ring)
    float topv = B[(ttr + 0)  * P + (1 + ttc + ln)];  // global row r0-1
    float botv = B[(ttr + 17) * P + (1 + ttc + ln)];  // global row r0+16
    acc[0] += (nh == 0) ? hy * topv : 0.0f;   // M=0  -> VGPR0 lanes 0-15
    acc[7] += (nh == 1) ? hy * botv : 0.0f;   // M=15 -> VGPR7 lanes 16-31
#pragma unroll
    for (int v = 0; v < 8; ++v) {
      int rowl = v + 8 * nh;                  // C layout: M = v + 8*nh, N = ln
      float lv = B[(1 + ttr + rowl) * P + (ttc + 0)];   // col c0-1
      float rv = B[(1 + ttr + rowl) * P + (ttc + 17)];  // col c0+16
      acc[v] += (ln == 0)  ? hx * lv : 0.0f;
      acc[v] += (ln == 15) ? hx * rv : 0.0f;
    }

    float* dp = d + (size_t)(R0 + ttr) * IMG + (C0 + ttc);
#pragma unroll
    for (int v = 0; v < 8; ++v)
      dp[(v + 8 * nh) * IMG + ln] = acc[v];
  }
}

extern "C" void kernel_launch(void* const* d_in, const int* in_sizes, int n_in,
                              void* d_out, int out_size, void* d_ws, size_t ws_size,
                              hipStream_t stream) {
  const float* x = (const float*)d_in[0];       // (16,2,1024,1024) f32
  const float* w = (const float*)d_in[1];       // (2,1,3,3) f32
  float* out = (float*)d_out;
  float* tmp = (float*)d_ws;                    // ping buffer (128 MB)

  int planes = in_sizes[0] / (IMG * IMG);       // 32

  dim3 grid(IMG / TILE, IMG / TILE, planes);
  dim3 block(256);

  const int T = 20;                             // reference time_steps
  const int L = T / FUSE;                       // 5 launches, 4 steps each
  const float* src = x;
  for (int s = 1; s <= L; ++s) {
    float* dst = (s & 1) ? out : tmp;           // launch 5 (odd) lands in d_out
    diffusion_fused4_wmma<<<grid, block, 0, stream>>>(src, w, dst);
    src = dst;
  }
}